// LightGCNModel_1056561955362
// MI455X (gfx1250) — compile-verified
//
#include <hip/hip_runtime.h>
#include <math.h>

typedef __attribute__((ext_vector_type(2))) float v2f;
typedef __attribute__((ext_vector_type(8))) float v8f;

#define DIM 64
#define NLAYERS 5

// ---------------- utility kernels ----------------

__global__ void zero_f32(float* __restrict__ p, int n) {
  int i = blockIdx.x * blockDim.x + threadIdx.x;
  if (i < n) p[i] = 0.0f;
}

// deg[col[e]] += 1   (target-degree, matching gcn_norm without self loops)
__global__ void deg_kernel(const int* __restrict__ col, float* __restrict__ deg, int E) {
  int e = blockIdx.x * blockDim.x + threadIdx.x;
  if (e < E) unsafeAtomicAdd(&deg[col[e]], 1.0f);
}

// deg -> deg^{-1/2} in place, with reference's where(deg>0, rsqrt(max(deg,1)), 0)
__global__ void dinv_kernel(float* __restrict__ deg, int n) {
  int i = blockIdx.x * blockDim.x + threadIdx.x;
  if (i < n) {
    float d = deg[i];
    deg[i] = (d > 0.0f) ? rsqrtf(fmaxf(d, 1.0f)) : 0.0f;
  }
}

__global__ void norm_kernel(const int* __restrict__ row, const int* __restrict__ col,
                            const float* __restrict__ dinv, float* __restrict__ nrm, int E) {
  int e = blockIdx.x * blockDim.x + threadIdx.x;
  if (e < E) nrm[e] = dinv[row[e]] * dinv[col[e]];
}

// x = emb ; out = alpha * emb   (layer-0 term)
__global__ void init_kernel(const float* __restrict__ emb, float* __restrict__ x,
                            float* __restrict__ out, float alpha, int n) {
  int i = blockIdx.x * blockDim.x + threadIdx.x;
  if (i < n) {
    float v = emb[i];
    x[i] = v;
    out[i] = alpha * v;
  }
}

// out += alpha * x
__global__ void axpy_kernel(const float* __restrict__ x, float* __restrict__ out,
                            float alpha, int n) {
  int i = blockIdx.x * blockDim.x + threadIdx.x;
  if (i < n) out[i] += alpha * x[i];
}

// xn[col[e], :] += norm[e] * x[row[e], :]
// 16 lanes per edge, each lane owns a float4 (4 dims): 1 coalesced b128 load +
// 4 hardware fp32 L2 atomics. 51.2MB table is L2-resident on MI455X (192MB L2).
__global__ void scatter_kernel(const int* __restrict__ row, const int* __restrict__ col,
                               const float* __restrict__ nrm, const float* __restrict__ x,
                               float* __restrict__ xn, int E) {
  int t = blockIdx.x * blockDim.x + threadIdx.x;
  int e = t >> 4;
  if (e >= E) return;
  int d = (t & 15) * 4;
  int r = row[e];
  int c = col[e];
  float w = nrm[e];
  const float4 v = *reinterpret_cast<const float4*>(x + (size_t)r * DIM + d);
  float* dst = xn + (size_t)c * DIM + d;
  unsafeAtomicAdd(dst + 0, w * v.x);
  unsafeAtomicAdd(dst + 1, w * v.y);
  unsafeAtomicAdd(dst + 2, w * v.z);
  unsafeAtomicAdd(dst + 3, w * v.w);
}

// ---------------- WMMA score kernel ----------------
// One wave handles 16 edges. D = A * B with
//   A[m,k]      = out[row[e0+m], k]   (16 x 64, fp32)
//   B[k,n]      = out[col[e0+n], k]   (64 x 16, fp32)
// so D[n,n] = score of edge e0+n. Accumulated as 16 chained
// V_WMMA_F32_16X16X4_F32 ops (K=4 each).
//
// f32 A(16x4) lane layout: lane L holds {A[L&15, kh], A[L&15, kh+1]}, kh=2*(L>>4)
// f32 B(4x16) lane layout: lane L holds {B[kh, L&15], B[kh+1, L&15]}
// => both operands are aligned float2 loads from the out table.
// f32 D(16x16): VGPR v, lanes 0-15 -> (M=v, N=lane); lanes 16-31 -> (M=v+8, N=lane-16).
// Diagonal: lanes 0-7 hold D[n,n] (n=lane) in acc[lane]; lanes 24-31 hold
// D[n,n] (n=lane-16) in acc[lane-24].
__global__ void score_wmma(const int* __restrict__ row, const int* __restrict__ col,
                           const float* __restrict__ out, float* __restrict__ scores) {
  int tid  = blockIdx.x * blockDim.x + threadIdx.x;
  int lane = tid & 31;
  size_t e0 = (size_t)(tid >> 5) * 16;   // grid sized so every wave is full & in range

  int n  = lane & 15;
  int kh = (lane >> 4) * 2;
  int r = row[e0 + n];
  int c = col[e0 + n];
  const float* pa = out + (size_t)r * DIM + kh;
  const float* pb = out + (size_t)c * DIM + kh;

  v8f acc = {};
#pragma unroll
  for (int kc = 0; kc < 16; ++kc) {
    v2f a = *reinterpret_cast<const v2f*>(pa + kc * 4);
    v2f b = *reinterpret_cast<const v2f*>(pb + kc * 4);
    acc = __builtin_amdgcn_wmma_f32_16x16x4_f32(
        /*neg_a=*/false, a, /*neg_b=*/false, b,
        /*c_mod=*/(short)0, acc, /*reuse_a=*/false, /*reuse_b=*/false);
  }

  // extract diagonal element without dynamic VGPR indexing
  int k = lane & 7;
  float val = acc[0];
  if (k == 1) val = acc[1];
  if (k == 2) val = acc[2];
  if (k == 3) val = acc[3];
  if (k == 4) val = acc[4];
  if (k == 5) val = acc[5];
  if (k == 6) val = acc[6];
  if (k == 7) val = acc[7];

  bool writer = (lane >> 4) == ((lane & 15) >> 3);
  if (writer) scores[e0 + n] = val;
}

// scalar tail (edges not covered by full-wave WMMA groups); also a correctness
// fallback path for arbitrary E.
__global__ void score_tail(const int* __restrict__ row, const int* __restrict__ col,
                           const float* __restrict__ out, float* __restrict__ scores,
                           int start, int E) {
  int e = start + blockIdx.x * blockDim.x + threadIdx.x;
  if (e >= E) return;
  const float* a = out + (size_t)row[e] * DIM;
  const float* b = out + (size_t)col[e] * DIM;
  float s = 0.0f;
#pragma unroll
  for (int d = 0; d < DIM; d += 4) {
    float4 va = *reinterpret_cast<const float4*>(a + d);
    float4 vb = *reinterpret_cast<const float4*>(b + d);
    s += va.x * vb.x + va.y * vb.y + va.z * vb.z + va.w * vb.w;
  }
  scores[e] = s;
}

// ---------------- launcher ----------------

extern "C" void kernel_launch(void* const* d_in, const int* in_sizes, int n_in,
                              void* d_out, int out_size, void* d_ws, size_t ws_size,
                              hipStream_t stream) {
  const int*   edge = (const int*)d_in[0];     // (2, E) int32
  const float* emb  = (const float*)d_in[1];   // (N, 64) fp32

  const int E  = in_sizes[0] / 2;
  const int NV = in_sizes[1];                  // N * 64
  const int N  = NV / DIM;

  const int* row = edge;
  const int* col = edge + E;
  float* scores  = (float*)d_out;

  // workspace layout (fp32): dinv | norm | x0 | x1 | out
  float* ws = (float*)d_ws;
  size_t off = 0;
  float* dinv = ws + off; off += (((size_t)N  + 255) & ~(size_t)255);
  float* nrm  = ws + off; off += (((size_t)E  + 255) & ~(size_t)255);
  float* x0   = ws + off; off += (size_t)NV;
  float* x1   = ws + off; off += (size_t)NV;
  float* oemb = ws + off;

  const int B = 256;                           // 8 wave32 per block
  const float alpha = 1.0f / (float)(NLAYERS + 1);

  zero_f32 <<<(N  + B - 1) / B, B, 0, stream>>>(dinv, N);
  deg_kernel <<<(E + B - 1) / B, B, 0, stream>>>(col, dinv, E);
  dinv_kernel<<<(N + B - 1) / B, B, 0, stream>>>(dinv, N);
  norm_kernel<<<(E + B - 1) / B, B, 0, stream>>>(row, col, dinv, nrm, E);
  init_kernel<<<(NV + B - 1) / B, B, 0, stream>>>(emb, x0, oemb, alpha, NV);

  float* xc = x0;
  float* xn = x1;
  for (int l = 0; l < NLAYERS; ++l) {
    zero_f32<<<(NV + B - 1) / B, B, 0, stream>>>(xn, NV);
    long T = (long)E * 16;                     // 16 lanes per edge
    scatter_kernel<<<(unsigned)((T + B - 1) / B), B, 0, stream>>>(row, col, nrm, xc, xn, E);
    axpy_kernel<<<(NV + B - 1) / B, B, 0, stream>>>(xn, oemb, alpha, NV);
    float* t = xc; xc = xn; xn = t;            // host-side swap, deterministic
  }

  // 128 edges per 256-thread block (16 edges per wave); only full blocks so
  // WMMA waves always run with EXEC all-ones.
  int Emain = (E / 128) * 128;
  if (Emain > 0)
    score_wmma<<<Emain / 128, B, 0, stream>>>(row, col, oemb, scores);
  int tail = E - Emain;
  if (tail > 0)
    score_tail<<<(tail + B - 1) / B, B, 0, stream>>>(row, col, oemb, scores, Emain, E);
}